// AxialDCNv4_45973329936717
// MI455X (gfx1250) — compile-verified
//
#include <hip/hip_runtime.h>

typedef __attribute__((ext_vector_type(2))) float v2f;
typedef __attribute__((ext_vector_type(4))) float v4f;
typedef __attribute__((ext_vector_type(8))) float v8f;

#define KK2   9
#define GRP   8
#define CG    16
#define CCH   128
#define KDIM  384    // C * 3 taps
#define MT    6      // 96 rows / 16
#define KSTEPS 96    // 384 / 4

// ---------------------------------------------------------------------------
// Pack (w_off || w_dyn || zeros) [96 x 384] into WMMA-A fragment order:
// apack[((mt*KSTEPS + ks)*32 + lane)*2 + t] = Wp[mt*16 + (lane&15)][ks*4 + ((lane>>4)*2) + t]
// Both 1x3 and 3x1 weights flatten identically: w[oc][c][tap] -> oc*384 + c*3 + tap.
// ---------------------------------------------------------------------------
__global__ void pack_weights_kernel(const float* __restrict__ woff,
                                    const float* __restrict__ wdyn,
                                    float* __restrict__ apack) {
    int i = blockIdx.x * blockDim.x + threadIdx.x;
    const int TOT = MT * KSTEPS * 64;
    if (i >= TOT) return;
    int t    = i & 1;
    int lane = (i >> 1) & 31;
    int rest = i >> 6;
    int ks   = rest % KSTEPS;
    int mt   = rest / KSTEPS;
    int row  = lane & 15;
    int kk   = (lane >> 4) * 2;
    int r    = mt * 16 + row;
    int k    = ks * 4 + kk + t;
    float v = 0.0f;
    if (r < 18)       v = woff[r * KDIM + k];
    else if (r < 90)  v = wdyn[(r - 18) * KDIM + k];
    apack[i] = v;
}

// ---------------------------------------------------------------------------
// Tiled NCHW -> NHWC transpose (32c x 32w tiles at fixed (b,h)).
// ---------------------------------------------------------------------------
__global__ __launch_bounds__(256) void transpose_nhwc_kernel(
    const float* __restrict__ x, float* __restrict__ xT, int B, int H, int W) {
    __shared__ float tile[32][33];
    const int nwt = W / 32;          // 5
    const int nct = CCH / 32;        // 4
    int blk = blockIdx.x;
    int wt  = blk % nwt;
    int ct  = (blk / nwt) % nct;
    int bh  = blk / (nwt * nct);
    int h   = bh % H;
    int b   = bh / H;
    int tx  = threadIdx.x & 31;
    int ty  = threadIdx.x >> 5;      // 0..7
#pragma unroll
    for (int r = 0; r < 4; ++r) {
        int c = ct * 32 + ty + r * 8;
        tile[ty + r * 8][tx] = x[(((size_t)b * CCH + c) * H + h) * W + wt * 32 + tx];
    }
    __syncthreads();
#pragma unroll
    for (int r = 0; r < 4; ++r) {
        int w = wt * 32 + ty + r * 8;
        xT[(((size_t)b * H + h) * W + w) * CCH + ct * 32 + tx] = tile[tx][ty + r * 8];
    }
}

// ---------------------------------------------------------------------------
// Fused offset+dynw conv as WMMA f32 GEMM.
// DIR==0: 1x3 kernel, pad (0,1) in w.  DIR==1: 3x1 kernel, pad (1,0) in h.
// Block: 192 threads = 6 waves; wave m handles M-tile m (16 output rows) and
// two 16-pixel N-tiles of a 32-pixel w-chunk at row (b,h).
// LDS staging uses CDNA5 async global->LDS copies (ASYNCcnt).
// ---------------------------------------------------------------------------
template <int DIR>
__global__ __launch_bounds__(192) void conv_wmma_kernel(
    const float* __restrict__ x,
    const float* __restrict__ apack,
    const float* __restrict__ bias_off,
    const float* __restrict__ bias_dyn,
    float* __restrict__ out_off,
    float* __restrict__ out_dyn,
    int B, int H, int W) {

    __shared__ float xs[3 * CCH * 32];   // 48 KB; DIR==0 uses first 128*34 floats

    const int nwc = W / 32;
    int blk = blockIdx.x;
    int wc  = blk % nwc;
    int bh  = blk / nwc;
    int h   = bh % H;
    int b   = bh / H;
    int w0  = wc * 32;

    int tid = threadIdx.x;
    const uint64_t xbase = (uint64_t)(uintptr_t)x;

    // ---- stage input tile into LDS via async copies; zero-fill padding ----
    if (DIR == 0) {
        const int TOT = CCH * 34;                 // [c][t], t = w-halo 0..33 -> w = w0-1+t
        for (int i = tid; i < TOT; i += 192) {
            int c = i / 34, t = i % 34;
            int wx = w0 - 1 + t;
            uint32_t ldsa = (uint32_t)(uintptr_t)(&xs[i]);
            if (wx >= 0 && wx < W) {
                int goff = (int)(((((size_t)b * CCH + c) * H + h) * W + wx) * sizeof(float));
                asm volatile("global_load_async_to_lds_b32 %0, %1, %2"
                             :: "v"(ldsa), "v"(goff), "s"(xbase) : "memory");
            } else {
                xs[i] = 0.0f;
            }
        }
    } else {
        const int TOT4 = (3 * CCH * 32) / 4;      // [j][c][t] in 4-float chunks
        for (int i4 = tid; i4 < TOT4; i4 += 192) {
            int i = i4 * 4;
            int t = i & 31;
            int c = (i >> 5) & 127;
            int j = i >> 12;
            int hy = h - 1 + j;
            uint32_t ldsa = (uint32_t)(uintptr_t)(&xs[i]);
            if (hy >= 0 && hy < H) {
                int goff = (int)(((((size_t)b * CCH + c) * H + hy) * W + w0 + t) * sizeof(float));
                asm volatile("global_load_async_to_lds_b128 %0, %1, %2"
                             :: "v"(ldsa), "v"(goff), "s"(xbase) : "memory");
            } else {
                *(v4f*)(&xs[i]) = (v4f){0.0f, 0.0f, 0.0f, 0.0f};
            }
        }
    }
    asm volatile("s_wait_asynccnt 0" ::: "memory");
    __syncthreads();

    int lane = tid & 31;
    int mt   = tid >> 5;                 // wave index == M-tile
    int n    = lane & 15;                // N column within tile / A row
    int kk   = (lane >> 4) * 2;          // K sub-offset for this half-wave

    const v2f* __restrict__ ap = (const v2f*)apack;
    int aofs = mt * KSTEPS * 32 + lane;

    v8f acc0 = {};
    v8f acc1 = {};

    // incremental (c,j) decomposition of k = ks*4 + kk (+1) with k = c*3 + j
    int c0 = 0,            j0 = kk;                  // kk in {0,2}
    int c1 = (kk + 1) / 3, j1 = (kk + 1) % 3;

    for (int ks = 0; ks < KSTEPS; ++ks) {
        v2f a = ap[aofs + ks * 32];

        float bx0, by0, bx1, by1;
        if (DIR == 0) {
            int u0 = c0 * 34 + j0 + n;
            int u1 = c1 * 34 + j1 + n;
            bx0 = xs[u0];       by0 = xs[u1];
            bx1 = xs[u0 + 16];  by1 = xs[u1 + 16];
        } else {
            int u0 = (j0 * CCH + c0) * 32 + n;
            int u1 = (j1 * CCH + c1) * 32 + n;
            bx0 = xs[u0];       by0 = xs[u1];
            bx1 = xs[u0 + 16];  by1 = xs[u1 + 16];
        }
        v2f b0 = {bx0, by0};
        v2f b1 = {bx1, by1};

        acc0 = __builtin_amdgcn_wmma_f32_16x16x4_f32(false, a, false, b0,
                                                     (short)0, acc0, false, false);
        acc1 = __builtin_amdgcn_wmma_f32_16x16x4_f32(false, a, false, b1,
                                                     (short)0, acc1, false, false);

        c0 += 1; j0 += 1; if (j0 >= 3) { j0 -= 3; c0 += 1; }
        c1 += 1; j1 += 1; if (j1 >= 3) { j1 -= 3; c1 += 1; }
    }

    // ---- epilogue: D layout -> global; rows >= 90 are padding ----
    int mb = (lane >> 4) * 8;
#pragma unroll
    for (int r = 0; r < 8; ++r) {
        int m  = r + mb;
        int oc = mt * 16 + m;
        if (oc < 90) {
            float bia;
            float* dst;
            if (oc < 18) {
                bia = bias_off[oc];
                dst = out_off + (((size_t)b * 18 + oc) * H + h) * W;
            } else {
                bia = bias_dyn[oc - 18];
                dst = out_dyn + (((size_t)b * 72 + (oc - 18)) * H + h) * W;
            }
            dst[w0 + n]      = acc0[r] + bia;
            dst[w0 + 16 + n] = acc1[r] + bia;
        }
    }
}

// ---------------------------------------------------------------------------
// Modulated deformable aggregation (DCNv4 style), NHWC source.
// One thread per (b, g, h, w): 9 taps, masked bilinear; the 16 group channels
// are contiguous in NHWC -> 4x b128 loads per corner.
// OUT_NHWC==1: write NHWC (feeds next deform pass). ==0: write NCHW (final).
// ---------------------------------------------------------------------------
template <int OUT_NHWC>
__global__ __launch_bounds__(256) void deform_agg_kernel(
    const float* __restrict__ srcT,  // [B,H,W,128]
    const float* __restrict__ off,   // [B,18,H,W]  ch 0..8 = dy_k, 9..17 = dx_k
    const float* __restrict__ dynw,  // [B,72,H,W]  ch g*9+k
    float* __restrict__ out,
    int B, int H, int W) {

    int idx = blockIdx.x * blockDim.x + threadIdx.x;
    int total = B * GRP * H * W;
    if (idx >= total) return;

    int w = idx % W;
    int h = (idx / W) % H;
    int g = (idx / (W * H)) % GRP;
    int b = idx / (W * H * GRP);

    size_t HW = (size_t)H * W;
    size_t hw = (size_t)h * W + w;
    const float* offb = off  + (size_t)b * 18 * HW + hw;
    const float* dynb = dynw + (size_t)b * 72 * HW + hw;
    const float* srcb = srcT + (size_t)b * HW * CCH + g * CG;

    v4f acc[4];
#pragma unroll
    for (int q = 0; q < 4; ++q) acc[q] = (v4f){0.0f, 0.0f, 0.0f, 0.0f};

#pragma unroll
    for (int k = 0; k < KK2; ++k) {
        float dy = offb[(size_t)k * HW];
        float dx = offb[(size_t)(9 + k) * HW];
        float m  = dynb[(size_t)(g * 9 + k) * HW];

        float py = (float)(h - 1 + k / 3) + dy;
        float px = (float)(w - 1 + k % 3) + dx;
        float y0f = floorf(py), x0f = floorf(px);
        float fy = py - y0f,    fx = px - x0f;
        int y0 = (int)y0f, x0 = (int)x0f;
        int y1 = y0 + 1,   x1 = x0 + 1;

        bool vy0 = (y0 >= 0) & (y0 < H);
        bool vy1 = (y1 >= 0) & (y1 < H);
        bool vx0 = (x0 >= 0) & (x0 < W);
        bool vx1 = (x1 >= 0) & (x1 < W);

        float w00 = (vy0 & vx0) ? m * (1.0f - fy) * (1.0f - fx) : 0.0f;
        float w01 = (vy0 & vx1) ? m * (1.0f - fy) * fx           : 0.0f;
        float w10 = (vy1 & vx0) ? m * fy * (1.0f - fx)           : 0.0f;
        float w11 = (vy1 & vx1) ? m * fy * fx                    : 0.0f;

        int y0c = min(max(y0, 0), H - 1), y1c = min(max(y1, 0), H - 1);
        int x0c = min(max(x0, 0), W - 1), x1c = min(max(x1, 0), W - 1);

        const v4f* p00 = (const v4f*)(srcb + ((size_t)y0c * W + x0c) * CCH);
        const v4f* p01 = (const v4f*)(srcb + ((size_t)y0c * W + x1c) * CCH);
        const v4f* p10 = (const v4f*)(srcb + ((size_t)y1c * W + x0c) * CCH);
        const v4f* p11 = (const v4f*)(srcb + ((size_t)y1c * W + x1c) * CCH);

#pragma unroll
        for (int q = 0; q < 4; ++q)
            acc[q] += w00 * p00[q] + w01 * p01[q] + w10 * p10[q] + w11 * p11[q];
    }

    if (OUT_NHWC) {
        v4f* op = (v4f*)(out + ((size_t)b * HW + hw) * CCH + g * CG);
#pragma unroll
        for (int q = 0; q < 4; ++q) op[q] = acc[q];
    } else {
        float* ob = out + ((size_t)b * CCH + g * CG) * HW + hw;
#pragma unroll
        for (int c = 0; c < CG; ++c) ob[(size_t)c * HW] = acc[c >> 2][c & 3];
    }
}

// ---------------------------------------------------------------------------
// Launch: pack -> transpose -> 2x conv (WMMA) -> 2x deform_agg.
// ---------------------------------------------------------------------------
extern "C" void kernel_launch(void* const* d_in, const int* in_sizes, int n_in,
                              void* d_out, int out_size, void* d_ws, size_t ws_size,
                              hipStream_t stream) {
    const float* x      = (const float*)d_in[0];
    const float* w_hoff = (const float*)d_in[1];
    const float* b_hoff = (const float*)d_in[2];
    const float* w_hw   = (const float*)d_in[3];
    const float* b_hw   = (const float*)d_in[4];
    const float* w_voff = (const float*)d_in[5];
    const float* b_voff = (const float*)d_in[6];
    const float* w_vw   = (const float*)d_in[7];
    const float* b_vw   = (const float*)d_in[8];

    const int H = 160, W = 160;
    const int B = in_sizes[0] / (CCH * H * W);
    const size_t HW = (size_t)H * W;

    // workspace layout (floats)
    float* ws = (float*)d_ws;
    const size_t APACK = (size_t)MT * KSTEPS * 64;   // 36864
    float* apack_h = ws;
    float* apack_v = apack_h + APACK;
    float* off_h   = apack_v + APACK;
    float* dyn_h   = off_h  + (size_t)B * 18 * HW;
    float* off_v   = dyn_h  + (size_t)B * 72 * HW;
    float* dyn_v   = off_v  + (size_t)B * 18 * HW;
    float* xT      = dyn_v  + (size_t)B * 72 * HW;   // [B,H,W,128]
    float* out_hT  = xT     + (size_t)B * HW * CCH;  // [B,H,W,128]
    (void)ws_size; (void)n_in; (void)out_size;

    // pack weights into WMMA-A fragment order
    {
        int tot = (int)APACK;
        int blocks = (tot + 255) / 256;
        pack_weights_kernel<<<blocks, 256, 0, stream>>>(w_hoff, w_hw, apack_h);
        pack_weights_kernel<<<blocks, 256, 0, stream>>>(w_voff, w_vw, apack_v);
    }

    // NCHW -> NHWC transpose of x (for deform gathers)
    {
        int blocks = B * H * (W / 32) * (CCH / 32);
        transpose_nhwc_kernel<<<blocks, 256, 0, stream>>>(x, xT, B, H, W);
    }

    // fused offset+dynw convolutions (WMMA GEMM, reads NCHW x)
    {
        int blocks = B * H * (W / 32);
        conv_wmma_kernel<0><<<blocks, 192, 0, stream>>>(x, apack_h, b_hoff, b_hw,
                                                        off_h, dyn_h, B, H, W);
        conv_wmma_kernel<1><<<blocks, 192, 0, stream>>>(x, apack_v, b_voff, b_vw,
                                                        off_v, dyn_v, B, H, W);
    }

    // deformable aggregation: horizontal on xT (NHWC->NHWC), vertical -> NCHW out
    {
        int tot = B * GRP * H * W;
        int blocks = (tot + 255) / 256;
        deform_agg_kernel<1><<<blocks, 256, 0, stream>>>(xT, off_h, dyn_h, out_hT,
                                                         B, H, W);
        deform_agg_kernel<0><<<blocks, 256, 0, stream>>>(out_hT, off_v, dyn_v,
                                                         (float*)d_out, B, H, W);
    }
}